// IPAFluxAttnProcessor2_0_6940667150622
// MI455X (gfx1250) — compile-verified
//
#include <hip/hip_runtime.h>
#include <hip/hip_bf16.h>
#include <math.h>

typedef __attribute__((ext_vector_type(2))) float v2f;
typedef __attribute__((ext_vector_type(4))) float v4f;
typedef __attribute__((ext_vector_type(8))) float v8f;

// D = A(16x4 f32) * B(4x16 f32) + C(16x16 f32)  -- native CDNA5 fp32 WMMA
#define WMMA_F32(A, B, C) \
  __builtin_amdgcn_wmma_f32_16x16x4_f32(false, (A), false, (B), (short)0, (C), false, false)

static constexpr int B_  = 2;
static constexpr int H_  = 24;
static constexpr int LQ  = 4608;
static constexpr int D_  = 128;
static constexpr int LIP = 16;
static constexpr int C_  = 4096;
static constexpr int HID = H_ * D_;   // 3072

// ---------------------------------------------------------------------------
// Kernel 1: k/v projection.  k[b,l,h] = sum_c emb[b,l,c] * W[h,c]
// Grid (192 n-tiles, 2 kv), 256 thr = 8 waves. Waves split the C contraction;
// each wave holds one 16x16 fp32 WMMA accumulator per batch (weights loaded
// once, shared across batches). LDS reduction across waves.
// Output kraw: [kv][b][l=16][hid=3072]
// ---------------------------------------------------------------------------
__global__ __launch_bounds__(256) void kv_proj_kernel(
    const float* __restrict__ emb,   // [B,16,4096]
    const float* __restrict__ Wk,    // [3072,4096]
    const float* __restrict__ Wv,    // [3072,4096]
    float* __restrict__ kraw)
{
  __shared__ float red[8 * 2 * 8 * 32];       // [wave][b][vgpr][lane] = 16 KB
  const int lane = threadIdx.x & 31;
  const int wv   = threadIdx.x >> 5;          // 0..7
  const int l16  = lane & 15;
  const int half = lane >> 4;
  const int nb   = blockIdx.x;                // 0..191 (16-wide output tile)
  const int kv   = blockIdx.y;                // 0:K 1:V
  const float* W = kv ? Wv : Wk;

  // A lane map: A[l16][2*half + j] ; B lane map: B[2*half + j][l16]
  const float* wrow  = W   + (size_t)(nb * 16 + l16) * C_ + 2 * half;
  const float* a0row = emb + (size_t)(0 * LIP + l16) * C_ + 2 * half;
  const float* a1row = emb + (size_t)(1 * LIP + l16) * C_ + 2 * half;

  v8f acc0 = {};
  v8f acc1 = {};
  const int c_begin = wv * (C_ / 8);
  const int c_end   = c_begin + (C_ / 8);
  for (int c0 = c_begin; c0 < c_end; c0 += 4) {
    v2f bvec = *reinterpret_cast<const v2f*>(wrow + c0);   // weight slice (shared)
    v2f a0   = *reinterpret_cast<const v2f*>(a0row + c0);  // batch 0
    v2f a1   = *reinterpret_cast<const v2f*>(a1row + c0);  // batch 1
    acc0 = WMMA_F32(a0, bvec, acc0);
    acc1 = WMMA_F32(a1, bvec, acc1);
  }

  #pragma unroll
  for (int r = 0; r < 8; ++r) {
    red[((wv * 2 + 0) * 8 + r) * 32 + lane] = acc0[r];
    red[((wv * 2 + 1) * 8 + r) * 32 + lane] = acc1[r];
  }
  __syncthreads();

  if (wv < 2) {                                // wave0 -> b=0, wave1 -> b=1
    const int b = wv;
    #pragma unroll
    for (int r = 0; r < 8; ++r) {
      float s = 0.f;
      #pragma unroll
      for (int w = 0; w < 8; ++w) s += red[((w * 2 + b) * 8 + r) * 32 + lane];
      const int row = r + 8 * half;            // C/D layout: rows r / r+8
      const int col = nb * 16 + l16;
      kraw[((size_t)((kv * 2 + b) * LIP + row)) * HID + col] = s;
    }
  }
}

// ---------------------------------------------------------------------------
// Kernel 2: per-head RMSNorm over D=128 (no affine). One wave per row.
// knorm layout: [kv][b][h][l=16][d=128]  (row-major per head -> direct
// B-matrix float2 loads in the attention kernel)
// ---------------------------------------------------------------------------
__global__ __launch_bounds__(256) void rms_kernel(
    const float* __restrict__ kraw,   // [kv][b][16][3072]
    float* __restrict__ knorm)
{
  const int lane = threadIdx.x & 31;
  const int wv   = threadIdx.x >> 5;
  const int idx  = blockIdx.x * 8 + wv;        // 0..1535 = (kv,b,h,l)
  const int l    = idx & 15;
  const int h    = (idx >> 4) % H_;
  const int b    = (idx / (16 * H_)) & 1;
  const int kv   = idx / (16 * H_ * 2);

  const float* src = kraw + ((size_t)((kv * 2 + b) * LIP + l)) * HID + h * D_ + lane * 4;
  v4f x = *reinterpret_cast<const v4f*>(src);
  float ss = x.x * x.x + x.y * x.y + x.z * x.z + x.w * x.w;
  #pragma unroll
  for (int m = 16; m >= 1; m >>= 1) ss += __shfl_xor(ss, m, 32);
  const float scale = rsqrtf(ss * (1.0f / D_) + 1e-5f);
  x *= scale;
  float* dst = knorm + ((size_t)(((kv * 2 + b) * H_ + h) * LIP + l)) * D_ + lane * 4;
  *reinterpret_cast<v4f*>(dst) = x;
}

// ---------------------------------------------------------------------------
// Kernel 3: attention. Block = (b, h, 4 q-tiles), 128 thr. K,V staged in LDS.
// Per wave: Q tile (16x128) x K^T (128x16) via 32 WMMAs -> 16x16 scores,
// softmax with half-wave shfl butterflies, attn -> per-wave LDS patch
// (re-enter in A layout), attn(16x16) x V(16x128) via 8x4 WMMAs.
// out: [B, Lq, H*D]
// ---------------------------------------------------------------------------
__global__ __launch_bounds__(128) void attn_kernel(
    const float* __restrict__ query,  // [B,H,4608,128]
    const float* __restrict__ knorm,  // [kv][b][h][16][128]
    float* __restrict__ out)
{
  __shared__ float Ksm[LIP * D_];     // 8 KB
  __shared__ float Vsm[LIP * D_];     // 8 KB
  __shared__ float Attn[4][LIP * 16]; // 4 KB, per-wave patches

  const int lane = threadIdx.x & 31;
  const int wv   = threadIdx.x >> 5;  // 0..3
  const int l16  = lane & 15;
  const int half = lane >> 4;
  const int h    = blockIdx.y;
  const int b    = blockIdx.z;

  const float* Kg = knorm + ((size_t)((0 * 2 + b) * H_ + h)) * (LIP * D_);
  const float* Vg = knorm + ((size_t)((1 * 2 + b) * H_ + h)) * (LIP * D_);
  for (int i = threadIdx.x; i < LIP * D_ / 4; i += 128) {
    reinterpret_cast<v4f*>(Ksm)[i] = reinterpret_cast<const v4f*>(Kg)[i];
    reinterpret_cast<v4f*>(Vsm)[i] = reinterpret_cast<const v4f*>(Vg)[i];
  }
  __syncthreads();

  const int q0 = (blockIdx.x * 4 + wv) * 16;
  const float* qrow =
      query + ((size_t)((b * H_ + h) * LQ + q0 + l16)) * D_ + 2 * half;

  // scores = Q . K^T
  v8f sc = {};
  for (int c0 = 0; c0 < D_; c0 += 4) {
    v2f a  = *reinterpret_cast<const v2f*>(qrow + c0);
    v2f bk = *reinterpret_cast<const v2f*>(&Ksm[l16 * D_ + c0 + 2 * half]);
    sc = WMMA_F32(a, bk, sc);
  }

  // softmax over 16 keys. Row M=r is lanes 0-15 of vgpr r, row r+8 lanes 16-31;
  // butterflies with masks 1,2,4,8 reduce each 16-lane half independently.
  const float qscale = 0.08838834764831845f;   // 128^-0.5
  float* aw = Attn[wv];
  #pragma unroll
  for (int r = 0; r < 8; ++r) {
    float v  = sc[r] * qscale;
    float mx = v;
    #pragma unroll
    for (int m = 8; m >= 1; m >>= 1) mx = fmaxf(mx, __shfl_xor(mx, m, 32));
    float e  = __expf(v - mx);
    float sm = e;
    #pragma unroll
    for (int m = 8; m >= 1; m >>= 1) sm += __shfl_xor(sm, m, 32);
    aw[(r + 8 * half) * 16 + l16] = e / sm;    // row-major 16x16 patch
  }
  // same-wave LDS RAW: hardware DScnt ordering handled by compiler waits.

  // out = attn . V    (8 column tiles of 16, K=16 -> 4 WMMAs each)
  #pragma unroll 1
  for (int nb = 0; nb < 8; ++nb) {
    v8f o = {};
    #pragma unroll
    for (int kc = 0; kc < 4; ++kc) {
      v2f a = *reinterpret_cast<const v2f*>(&aw[l16 * 16 + kc * 4 + 2 * half]);
      v2f bv;
      bv.x = Vsm[(kc * 4 + 2 * half + 0) * D_ + nb * 16 + l16];
      bv.y = Vsm[(kc * 4 + 2 * half + 1) * D_ + nb * 16 + l16];
      o = WMMA_F32(a, bv, o);
    }
    #pragma unroll
    for (int r = 0; r < 8; ++r) {
      out[((size_t)b * LQ + q0 + r + 8 * half) * HID + h * D_ + nb * 16 + l16] = o[r];
    }
  }
}

// ---------------------------------------------------------------------------
extern "C" void kernel_launch(void* const* d_in, const int* in_sizes, int n_in,
                              void* d_out, int out_size, void* d_ws, size_t ws_size,
                              hipStream_t stream) {
  const float* query = (const float*)d_in[0];  // [2,24,4608,128]
  const float* emb   = (const float*)d_in[1];  // [2,16,4096]
  // d_in[2] = t (unused by reference)
  const float* Wk    = (const float*)d_in[3];  // [3072,4096]
  const float* Wv    = (const float*)d_in[4];  // [3072,4096]
  // d_in[5] = num_heads (constant 24)
  float* out = (float*)d_out;                  // [2,4608,3072]

  float* kraw  = (float*)d_ws;                           // 2*2*16*3072 floats
  float* knorm = kraw + (size_t)2 * B_ * LIP * HID;      // same size

  kv_proj_kernel<<<dim3(HID / 16, 2), 256, 0, stream>>>(emb, Wk, Wv, kraw);
  rms_kernel<<<dim3((2 * B_ * H_ * LIP) / 8), 256, 0, stream>>>(kraw, knorm);
  attn_kernel<<<dim3(LQ / 64, H_, B_), 128, 0, stream>>>(query, knorm, out);
}